// NeuralRibosome_52896817218072
// MI455X (gfx1250) — compile-verified
//
#include <hip/hip_runtime.h>

// IFS fractal iteration for MI455X (gfx1250, wave32).
//
// 1 tiny setup kernel (normalize probs + pixel-space affine coefficients),
// then 20 dependent full-canvas kernels (ping-pong in d_ws, final one writes
// d_out and adds base_value). The 4MB canvas is L2-resident (192MB L2);
// the kernel is gather-bound, so the bilinear tap is fully branch-free
// (clamped unconditional loads + cndmask'd weights, no EXEC churn).
//
// The 16-transform weighted reduction uses V_WMMA_F32_16X16X4_F32:
// A = p replicated over M rows, B = bilinear samples (K=4 transforms x
// N=16 pixels per call, 4 calls accumulate T=16). All D rows are identical,
// so lanes 0..15 read acc[0] as the 16 pixel outputs.

typedef float v2f __attribute__((ext_vector_type(2)));
typedef float v8f __attribute__((ext_vector_type(8)));

#define HW 1024
#define TT 16
#define ITERS 20
#define SEGS 4   // 16-pixel segments per wave (64 pixels per wave-task)

// ---- setup: p[t] = probs[t]/sum(probs); pixel-space affine coefficients ----
// ix = t00*w + t01*h + Cx,  Cx = -511.5*(t00+t01) + 512*t02 + 511.5
// iy = t10*w + t11*h + Cy,  Cy = -511.5*(t10+t11) + 512*t12 + 511.5
// (exact algebraic rewrite of affine_grid + unnormalize, W=H=1024,
//  align_corners=False)
__global__ void ifs_setup_kernel(const float* __restrict__ theta,
                                 const float* __restrict__ probs,
                                 float* __restrict__ coef) {
  int t = threadIdx.x;
  if (t < TT) {
    float s = 0.f;
    for (int j = 0; j < TT; ++j) s += probs[j];
    float p = probs[t] / s;
    const float* th = theta + t * 6;
    float t00 = th[0], t01 = th[1], t02 = th[2];
    float t10 = th[3], t11 = th[4], t12 = th[5];
    float4 cx, cy;
    cx.x = t00; cx.y = t01;
    cx.z = fmaf(512.f, t02, 511.5f) - 511.5f * (t00 + t01);
    cx.w = p;                       // weight rides in .w
    cy.x = t10; cy.y = t11;
    cy.z = fmaf(512.f, t12, 511.5f) - 511.5f * (t10 + t11);
    cy.w = 0.f;
    ((float4*)coef)[t * 2 + 0] = cx;
    ((float4*)coef)[t * 2 + 1] = cy;
  }
}

// Branch-free bilinear tap with zeros padding: clamp the gather addresses,
// zero the weights of out-of-range taps (== clip + where in the reference).
__device__ __forceinline__ float ifs_bsample(const float* __restrict__ src,
                                             float ix, float iy) {
  float x0f = floorf(ix), y0f = floorf(iy);
  float wx1 = ix - x0f, wy1 = iy - y0f;
  float wx0 = 1.f - wx1, wy0 = 1.f - wy1;
  int x0 = (int)x0f, y0 = (int)y0f;
  int x1 = x0 + 1, y1 = y0 + 1;
  wx0 = ((unsigned)x0 < (unsigned)HW) ? wx0 : 0.f;
  wx1 = ((unsigned)x1 < (unsigned)HW) ? wx1 : 0.f;
  wy0 = ((unsigned)y0 < (unsigned)HW) ? wy0 : 0.f;
  wy1 = ((unsigned)y1 < (unsigned)HW) ? wy1 : 0.f;
  int x0c = min(max(x0, 0), HW - 1), x1c = min(max(x1, 0), HW - 1);
  int y0c = min(max(y0, 0), HW - 1), y1c = min(max(y1, 0), HW - 1);
  int r0 = y0c << 10, r1 = y1c << 10;
  float v00 = src[r0 + x0c];
  float v10 = src[r0 + x1c];
  float v01 = src[r1 + x0c];
  float v11 = src[r1 + x1c];
  return wy0 * fmaf(wx0, v00, wx1 * v10) + wy1 * fmaf(wx0, v01, wx1 * v11);
}

// One IFS iteration. Each wave owns SEGS consecutive 16-pixel segments of one
// row. Lane L: pixel n = L&15 within a segment, transform-half khalf = L>>4.
// Per lane the 8 relevant transforms' coefficients are hoisted to registers
// with the row term pre-folded (R = B*h + C), so each sample needs only
// ix = fma(A, w, R).
__global__ __launch_bounds__(256) void ifs_iter_kernel(
    const float* __restrict__ src, float* __restrict__ dst,
    const float* __restrict__ coef, const float* __restrict__ basev) {
  const int lane = threadIdx.x & 31;
  const int wave = threadIdx.x >> 5;
  const int task = blockIdx.x * 8 + wave;   // 16384 tasks = 1024 rows * 16
  const int row = task >> 4;
  const int wbase0 = (task & 15) * (16 * SEGS);
  const float hf = (float)row;
  const int khalf = lane >> 4;              // 0 or 1
  const float lanex = (float)(lane & 15);

  // Hoist this lane's 8 transforms: i = kb*2 + v  ->  t = kb*4 + khalf*2 + v
  const float4* cp = (const float4*)coef;
  float Ax[8], Rx[8], Ay[8], Ry[8], Pw[8];
#pragma unroll
  for (int i = 0; i < 8; ++i) {
    int t = ((i >> 1) << 2) + (khalf << 1) + (i & 1);
    float4 cx = cp[t * 2 + 0];
    float4 cy = cp[t * 2 + 1];
    Ax[i] = cx.x; Rx[i] = fmaf(cx.y, hf, cx.z);
    Ay[i] = cy.x; Ry[i] = fmaf(cy.y, hf, cy.z);
    Pw[i] = cx.w;
  }

  const float base = basev ? basev[0] : 0.f;

  for (int s = 0; s < SEGS; ++s) {
    const int wbase = wbase0 + s * 16;
    const float wf = (float)wbase + lanex;

    v8f acc = {0.f, 0.f, 0.f, 0.f, 0.f, 0.f, 0.f, 0.f};
#pragma unroll
    for (int kb = 0; kb < 4; ++kb) {
      const int i0 = kb * 2, i1 = kb * 2 + 1;
      float ix0 = fmaf(Ax[i0], wf, Rx[i0]);
      float iy0 = fmaf(Ay[i0], wf, Ry[i0]);
      float ix1 = fmaf(Ax[i1], wf, Rx[i1]);
      float iy1 = fmaf(Ay[i1], wf, Ry[i1]);

      v2f a = {Pw[i0], Pw[i1]};            // A: p replicated across M rows
      v2f b = {ifs_bsample(src, ix0, iy0), // B: K x N sampled values
               ifs_bsample(src, ix1, iy1)};
      // D = A(16x4) * B(4x16) + C ; accumulates 4 transforms per call.
      acc = __builtin_amdgcn_wmma_f32_16x16x4_f32(
          false, a, false, b, (short)0, acc, false, false);
    }
    // D[0][n] sits in acc[0] of lanes 0..15 (all rows identical).
    if (lane < 16) dst[(size_t)row * HW + wbase + lane] = acc[0] + base;
  }
}

extern "C" void kernel_launch(void* const* d_in, const int* in_sizes, int n_in,
                              void* d_out, int out_size, void* d_ws,
                              size_t ws_size, hipStream_t stream) {
  (void)in_sizes; (void)n_in; (void)out_size; (void)ws_size;
  const float* canvas0 = (const float*)d_in[0];  // [1024*1024]
  const float* theta   = (const float*)d_in[1];  // [16,2,3]
  const float* probs   = (const float*)d_in[2];  // [16]
  const float* basev   = (const float*)d_in[3];  // [1]
  float* out = (float*)d_out;

  char* ws = (char*)d_ws;
  float* coef = (float*)ws;                                  // 16 * 8 floats
  float* bufA = (float*)(ws + 1024);                         // 4 MB
  float* bufB = (float*)(ws + 1024 + (size_t)HW * HW * 4);   // 4 MB

  ifs_setup_kernel<<<1, 32, 0, stream>>>(theta, probs, coef);

  const float* src = canvas0;
  float* pingpong[2] = {bufA, bufB};
  const int nblocks = (HW * (HW / (16 * SEGS))) / 8;         // 2048
  for (int it = 0; it < ITERS; ++it) {
    const bool last = (it == ITERS - 1);
    float* dst = last ? out : pingpong[it & 1];
    ifs_iter_kernel<<<nblocks, 256, 0, stream>>>(src, dst, coef,
                                                 last ? basev : nullptr);
    src = dst;
  }
}